// Net_18107582120129
// MI455X (gfx1250) — compile-verified
//
#include <hip/hip_runtime.h>
#include <hip/hip_bf16.h>
#include <math.h>

typedef __attribute__((ext_vector_type(2))) float v2f;
typedef __attribute__((ext_vector_type(8))) float v8f;

// fp32 tensor op: D(16x16) = A(16x4) * B(4x16) + C
__device__ __forceinline__ v8f wmma4(v2f a, v2f b, v8f c) {
    return __builtin_amdgcn_wmma_f32_16x16x4_f32(
        /*neg_a=*/false, a, /*neg_b=*/false, b,
        /*c_mod=*/(short)0, c, /*reuse_a=*/false, /*reuse_b=*/false);
}

// relu as a single v_med3_f32: median(x, 0, +huge) == max(x, 0)
__device__ __forceinline__ float relu1(float x) {
    return __builtin_amdgcn_fmed3f(x, 0.0f, 3.0e38f);
}

// Feature permutation making D-layout == next layer's B-layout:
// physical slot M = 16t + r + 8hf carries logical feature g(M).
__device__ __forceinline__ int gperm(int M) {
    return 4 * (M & 7) + (M >> 4) + 2 * ((M >> 3) & 1);
}

// Constant operands pre-swizzled to WMMA register layouts (g-permuted rows).
struct C5 {
    v2f a1[2];      // W1^T: 2 output tiles, one K-slice (K=3 padded to 4)
    v2f a2[2][8];   // W2^T: 2 output tiles x 8 K-slices
    v2f a3[8];      // W3^T: outputs at rows {0,1,8}, 8 K-slices
    v8f c1[2];      // b1 in g-permuted C layout
    v8f c2[2];      // b2
    v8f c3;         // b3 at rows {0,1,8}
};

// One MLP eval for 16 points. State is packed dual-half:
//   Y0 = (y0 | y2) across lane halves, Y1 = (y1 | 0).
// Output F0 = (k0 | k2), F1 = (k1 | 0). Zero cross-lane ops.
__device__ __forceinline__ void mlp3(const C5& W, float Y0, float Y1,
                                     float& F0, float& F1) {
    // ---- layer 1: B operand is the state itself ----
    v2f B1; B1.x = Y0; B1.y = Y1;
    v8f d0 = wmma4(W.a1[0], B1, W.c1[0]);
    v8f d1 = wmma4(W.a1[1], B1, W.c1[1]);
    // relu + pack: B-slice s of layer 2 is {d0[s], d1[s]} (g-permutation)
    v2f Hs[8];
#pragma unroll
    for (int s = 0; s < 8; ++s) {
        Hs[s].x = relu1(d0[s]);
        Hs[s].y = relu1(d1[s]);
    }
    // ---- layer 2: K=32 as 8 slices, bias in C, accumulate D->C ----
    v8f e0 = W.c2[0], e1 = W.c2[1];
#pragma unroll
    for (int s = 0; s < 8; ++s) {
        e0 = wmma4(W.a2[0][s], Hs[s], e0);
        e1 = wmma4(W.a2[1][s], Hs[s], e1);
    }
    v2f Gs[8];
#pragma unroll
    for (int s = 0; s < 8; ++s) {
        Gs[s].x = relu1(e0[s]);
        Gs[s].y = relu1(e1[s]);
    }
    // ---- layer 3: outputs land at rows {0,1,8} -> packed (k0|k2),(k1|0) ----
    v8f f0 = W.c3;
#pragma unroll
    for (int s = 0; s < 8; ++s) {
        f0 = wmma4(W.a3[s], Gs[s], f0);
    }
    F0 = f0[0];   // lanes 0-15: k0, lanes 16-31: k2
    F1 = f0[1];   // lanes 0-15: k1, lanes 16-31: 0 (row 9 is a zero row)
}

__global__ __launch_bounds__(256) void node_rk4_kernel(
    const float* __restrict__ x, const float* __restrict__ samples,
    const float* __restrict__ w1, const float* __restrict__ b1,
    const float* __restrict__ w2, const float* __restrict__ b2,
    const float* __restrict__ w3, const float* __restrict__ b3,
    const float* __restrict__ w_out, const float* __restrict__ b_out,
    float* __restrict__ out, int Btot) {

    const int lane = threadIdx.x & 31;
    const int lm = lane & 15;
    const int hf = lane >> 4;
    const int wave = blockIdx.x * (blockDim.x >> 5) + (threadIdx.x >> 5);
    const int p = wave * 16 + lm;               // this lane's point
    const int pc = p < Btot ? p : Btot - 1;     // clamped for loads
    const bool writer = (lane < 16) && (p < Btot);

    // ---- pre-swizzle weights (rows g-permuted so no runtime shuffles) ----
    C5 W;
#pragma unroll
    for (int t = 0; t < 2; ++t) {
        const int gm = gperm(16 * t + lm);      // logical hidden feature of row M
        {
            const int k0i = 2 * hf + 0, k1i = 2 * hf + 1;
            W.a1[t].x = (k0i < 3) ? w1[k0i * 32 + gm] : 0.0f;
            W.a1[t].y = (k1i < 3) ? w1[k1i * 32 + gm] : 0.0f;
        }
#pragma unroll
        for (int s = 0; s < 8; ++s) {
            const int ka = 4 * s + 2 * hf;      // logical in-feature (plain order)
            W.a2[t][s].x = w2[(ka + 0) * 32 + gm];
            W.a2[t][s].y = w2[(ka + 1) * 32 + gm];
        }
#pragma unroll
        for (int r = 0; r < 8; ++r) {
            const int gf = 4 * r + t + 2 * hf;  // g(16t + r + 8hf)
            W.c1[t][r] = b1[gf];
            W.c2[t][r] = b2[gf];
        }
    }
    // layer 3: output rows 0,1,8 hold out0,out1,out2; all other rows zero
    const int o3 = (lm == 0) ? 0 : (lm == 1) ? 1 : (lm == 8) ? 2 : -1;
#pragma unroll
    for (int s = 0; s < 8; ++s) {
        const int ka = 4 * s + 2 * hf;
        W.a3[s].x = (o3 >= 0) ? w3[(ka + 0) * 3 + o3] : 0.0f;
        W.a3[s].y = (o3 >= 0) ? w3[(ka + 1) * 3 + o3] : 0.0f;
    }
#pragma unroll
    for (int r = 0; r < 8; ++r) {
        const int M = r + 8 * hf;
        W.c3[r] = (M == 0) ? b3[0] : (M == 1) ? b3[1] : (M == 8) ? b3[2] : 0.0f;
    }

    // ---- time grid, output step indices, and a 64-bit hit bitmask ----
    const float maxT = samples[7];              // samples sorted ascending
    const float dt = maxT / 64.0f;
    const float dt6 = dt / 6.0f;
    const float hdt = 0.5f * dt;
    int sidx[8];
    unsigned long long hitmask = 0ull;
#pragma unroll
    for (int q = 0; q < 8; ++q) {
        int i = (int)rintf(samples[q] / dt) - 1;
        sidx[q] = i < 0 ? 0 : (i > 63 ? 63 : i);
        hitmask |= 1ull << sidx[q];
    }
    const float wo0 = w_out[0], wo1 = w_out[1], wo2 = w_out[2], bo = b_out[0];

    // ---- initial state in packed dual-half form ----
    float Y0 = x[pc * 3 + (hf ? 2 : 0)];        // y0 | y2
    float Y1 = hf ? 0.0f : x[pc * 3 + 1];       // y1 | 0

    // ---- 64 RK4 steps, 4 MLP evals each, all WMMA, zero ds ops ----
#pragma unroll 1
    for (int i = 0; i < 64; ++i) {
        float A0, A1, B0, B1, C0, C1, D0, D1;
        mlp3(W, Y0, Y1, A0, A1);
        mlp3(W, Y0 + hdt * A0, Y1 + hdt * A1, B0, B1);
        mlp3(W, Y0 + hdt * B0, Y1 + hdt * B1, C0, C1);
        mlp3(W, Y0 + dt * C0, Y1 + dt * C1, D0, D1);
        Y0 += dt6 * (A0 + 2.0f * B0 + 2.0f * C0 + D0);
        Y1 += dt6 * (A1 + 2.0f * B1 + 2.0f * C1 + D1);

        // single bit-test per step; taken 8 times total (wave-uniform)
        if ((hitmask >> i) & 1ull) {
            // bring y2 (lanes 16-31 of Y0) down to the low half
            const float y2lo = __shfl(Y0, lm | 16, 32);
            const float o = Y0 * wo0 + Y1 * wo1 + y2lo * wo2 + bo;
#pragma unroll
            for (int q = 0; q < 8; ++q) {
                if (sidx[q] == i && writer) {
                    out[(size_t)q * (size_t)Btot + (size_t)p] = o;
                }
            }
        }
    }
}

extern "C" void kernel_launch(void* const* d_in, const int* in_sizes, int n_in,
                              void* d_out, int out_size, void* d_ws, size_t ws_size,
                              hipStream_t stream) {
    const float* x       = (const float*)d_in[0];
    const float* samples = (const float*)d_in[1];
    const float* w1      = (const float*)d_in[2];
    const float* b1      = (const float*)d_in[3];
    const float* w2      = (const float*)d_in[4];
    const float* b2      = (const float*)d_in[5];
    const float* w3      = (const float*)d_in[6];
    const float* b3      = (const float*)d_in[7];
    const float* w_out   = (const float*)d_in[8];
    const float* b_out   = (const float*)d_in[9];
    float* out = (float*)d_out;

    const int Btot = in_sizes[0] / 3;           // 131072
    const int pointsPerBlock = 128;             // 8 waves x 16 points
    const int blocks = (Btot + pointsPerBlock - 1) / pointsPerBlock;
    node_rk4_kernel<<<blocks, 256, 0, stream>>>(
        x, samples, w1, b1, w2, b2, w3, b3, w_out, b_out, out, Btot);
}